// GAT_75067438399500
// MI455X (gfx1250) — compile-verified
//
#include <hip/hip_runtime.h>
#include <hip/hip_bf16.h>
#include <math.h>

#define NODES 50000
#define NEDGE 800000
#define FIN   256
#define NHEAD 4
#define DH    32
#define HID   128
#define FOUT  64
#define SLOPE 0.2f

typedef __attribute__((ext_vector_type(2))) float v2f;
typedef __attribute__((ext_vector_type(8))) float v8f;

// ---------------------------------------------------------------------------
// Dense GEMM  C[M,NCOL] = A[M,K] @ B[K,NCOL]  via V_WMMA_F32_16X16X4_F32.
// One wave computes one 16x16 tile of C. M must be a multiple of 16,
// K a multiple of 4, NCOL a multiple of 16 (all hold for this problem).
// A-frag layout (16x4 f32): lanes 0-15 -> K=k0,k0+1 ; lanes 16-31 -> K=k0+2,k0+3
// B-frag layout (4x16 f32): VGPR0 = rows k0 / k0+2, VGPR1 = rows k0+1 / k0+3
// C/D layout: VGPR r -> row r (lanes 0-15), row r+8 (lanes 16-31)
// ---------------------------------------------------------------------------
template <int K, int NCOL>
__global__ __launch_bounds__(32) void wmma_gemm_f32(const float* __restrict__ A,
                                                    const float* __restrict__ B,
                                                    float* __restrict__ C) {
    const int lane = threadIdx.x;
    const int l    = lane & 15;
    const int half = lane >> 4;
    const int row0 = blockIdx.x * 16;
    const int col0 = blockIdx.y * 16;

    const float* __restrict__ arow = A + (size_t)(row0 + l) * K;
    const float* __restrict__ bcol = B + col0 + l;

    v8f acc = {};
#pragma unroll 4
    for (int k0 = 0; k0 < K; k0 += 4) {
        const int ka = k0 + 2 * half;
        v2f a, b;
        a.x = arow[ka];
        a.y = arow[ka + 1];
        b.x = bcol[(size_t)ka * NCOL];
        b.y = bcol[(size_t)(ka + 1) * NCOL];
        acc = __builtin_amdgcn_wmma_f32_16x16x4_f32(
            /*neg_a=*/false, a, /*neg_b=*/false, b,
            /*c_mod=*/(short)0, acc, /*reuse_a=*/false, /*reuse_b=*/false);
    }

    float* __restrict__ crow = C + (size_t)(row0 + 8 * half) * NCOL + col0 + l;
#pragma unroll
    for (int r = 0; r < 8; ++r) crow[(size_t)r * NCOL] = acc[r];
}

// ---------------------------------------------------------------------------
// el[n,h] = feat[n,h,:] . al[h,:]   er likewise. idx = n*H + h, feat row-major.
// ---------------------------------------------------------------------------
__global__ void attn_scores(const float* __restrict__ feat,
                            const float* __restrict__ al,
                            const float* __restrict__ ar,
                            float* __restrict__ el, float* __restrict__ er,
                            int n, int heads, int d) {
    int idx = blockIdx.x * blockDim.x + threadIdx.x;
    if (idx >= n * heads) return;
    int h = idx % heads;
    const float* __restrict__ f  = feat + (size_t)idx * d;
    const float* __restrict__ vl = al + h * d;
    const float* __restrict__ vr = ar + h * d;
    float sl = 0.f, sr = 0.f;
    for (int i = 0; i < d; ++i) {
        float v = f[i];
        sl += v * vl[i];
        sr += v * vr[i];
    }
    el[idx] = sl;
    er[idx] = sr;
}

__device__ __forceinline__ float leaky(float x) { return x > 0.f ? x : SLOPE * x; }

// sign-split float atomic max (works with -inf initialization)
__device__ __forceinline__ void atomicMaxF(float* addr, float val) {
    if (val >= 0.f)
        atomicMax((int*)addr, __float_as_int(val));
    else
        atomicMin((unsigned int*)addr, __float_as_uint(val));
}

__global__ void fill_f32(float* __restrict__ p, long n, float v) {
    long i = (long)blockIdx.x * blockDim.x + threadIdx.x;
    if (i < n) p[i] = v;
}

// pass 1: segment max of leaky_relu(el[src]+er[dst]) over dst
__global__ void edge_max(const int* __restrict__ src, const int* __restrict__ dst,
                         const float* __restrict__ el, const float* __restrict__ er,
                         float* __restrict__ m, int ne, int H) {
    int t = blockIdx.x * blockDim.x + threadIdx.x;
    if (t >= ne * H) return;
    int e = t / H, h = t - e * H;
    int s = src[e], d = dst[e];
    float sc = leaky(el[s * H + h] + er[d * H + h]);
    atomicMaxF(&m[d * H + h], sc);
}

// pass 2: z[dst] += exp(score - m[dst])
__global__ void edge_expsum(const int* __restrict__ src, const int* __restrict__ dst,
                            const float* __restrict__ el, const float* __restrict__ er,
                            const float* __restrict__ m, float* __restrict__ z,
                            int ne, int H) {
    int t = blockIdx.x * blockDim.x + threadIdx.x;
    if (t >= ne * H) return;
    int e = t / H, h = t - e * H;
    int s = src[e], d = dst[e];
    float sc = leaky(el[s * H + h] + er[d * H + h]);
    atomicAdd(&z[d * H + h], __expf(sc - m[d * H + h]));
}

// pass 3: out[dst,h,:] += feat[src,h,:] * exp(score - m)/z
__global__ void edge_aggregate(const int* __restrict__ src, const int* __restrict__ dst,
                               const float* __restrict__ el, const float* __restrict__ er,
                               const float* __restrict__ m, const float* __restrict__ z,
                               const float* __restrict__ feat, float* __restrict__ out,
                               int ne, int H, int D) {
    int t = blockIdx.x * blockDim.x + threadIdx.x;
    if (t >= ne * H) return;
    int e = t / H, h = t - e * H;
    int s = src[e], d = dst[e];
    float sc = leaky(el[s * H + h] + er[d * H + h]);
    float a  = __expf(sc - m[d * H + h]) / z[d * H + h];
    const float* __restrict__ fs = feat + ((size_t)s * H + h) * D;
    float* __restrict__ od       = out + ((size_t)d * H + h) * D;
    for (int i = 0; i < D; ++i) atomicAdd(&od[i], fs[i] * a);
}

// h[i] = relu(agg[i] + b[i % F])
__global__ void bias_relu(const float* __restrict__ agg, const float* __restrict__ b,
                          float* __restrict__ h, long n, int F) {
    long i = (long)blockIdx.x * blockDim.x + threadIdx.x;
    if (i >= n) return;
    float v = agg[i] + b[i % F];
    h[i] = v > 0.f ? v : 0.f;
}

// out[i] += b[i % F]  (final bias, no activation)
__global__ void bias_add(float* __restrict__ out, const float* __restrict__ b,
                         long n, int F) {
    long i = (long)blockIdx.x * blockDim.x + threadIdx.x;
    if (i >= n) return;
    out[i] += b[i % F];
}

static inline int cdiv(long a, int b) { return (int)((a + b - 1) / b); }

extern "C" void kernel_launch(void* const* d_in, const int* in_sizes, int n_in,
                              void* d_out, int out_size, void* d_ws, size_t ws_size,
                              hipStream_t stream) {
    const float* x   = (const float*)d_in[0];
    const float* W1  = (const float*)d_in[1];
    const float* al1 = (const float*)d_in[2];
    const float* ar1 = (const float*)d_in[3];
    const float* b1  = (const float*)d_in[4];
    const float* W2  = (const float*)d_in[5];
    const float* al2 = (const float*)d_in[6];
    const float* ar2 = (const float*)d_in[7];
    const float* b2  = (const float*)d_in[8];
    const int*   src = (const int*)d_in[9];
    const int*   dst = (const int*)d_in[10];
    float*       out = (float*)d_out;

    // workspace carve-up (floats)
    float* feat1 = (float*)d_ws;                      // N*HID  (also reused as h1)
    float* agg1  = feat1 + (size_t)NODES * HID;       // N*HID
    float* el1   = agg1 + (size_t)NODES * HID;        // N*NHEAD
    float* er1   = el1 + (size_t)NODES * NHEAD;
    float* m1    = er1 + (size_t)NODES * NHEAD;
    float* z1    = m1 + (size_t)NODES * NHEAD;
    float* feat2 = z1 + (size_t)NODES * NHEAD;        // N*FOUT
    float* el2   = feat2 + (size_t)NODES * FOUT;      // N
    float* er2   = el2 + NODES;
    float* m2    = er2 + NODES;
    float* z2    = m2 + NODES;

    const int TB = 256;

    // ---------------- layer 1 ----------------
    wmma_gemm_f32<FIN, HID><<<dim3(NODES / 16, HID / 16), 32, 0, stream>>>(x, W1, feat1);

    attn_scores<<<cdiv((long)NODES * NHEAD, TB), TB, 0, stream>>>(feat1, al1, ar1, el1, er1,
                                                                  NODES, NHEAD, DH);

    fill_f32<<<cdiv((long)NODES * NHEAD, TB), TB, 0, stream>>>(m1, (long)NODES * NHEAD, -INFINITY);
    fill_f32<<<cdiv((long)NODES * NHEAD, TB), TB, 0, stream>>>(z1, (long)NODES * NHEAD, 0.f);
    fill_f32<<<cdiv((long)NODES * HID, TB), TB, 0, stream>>>(agg1, (long)NODES * HID, 0.f);

    edge_max<<<cdiv((long)NEDGE * NHEAD, TB), TB, 0, stream>>>(src, dst, el1, er1, m1, NEDGE, NHEAD);
    edge_expsum<<<cdiv((long)NEDGE * NHEAD, TB), TB, 0, stream>>>(src, dst, el1, er1, m1, z1,
                                                                  NEDGE, NHEAD);
    edge_aggregate<<<cdiv((long)NEDGE * NHEAD, TB), TB, 0, stream>>>(src, dst, el1, er1, m1, z1,
                                                                     feat1, agg1, NEDGE, NHEAD, DH);

    // h1 = relu(agg1 + b1), stored back into feat1 buffer
    bias_relu<<<cdiv((long)NODES * HID, TB), TB, 0, stream>>>(agg1, b1, feat1,
                                                              (long)NODES * HID, HID);

    // ---------------- layer 2 ----------------
    wmma_gemm_f32<HID, FOUT><<<dim3(NODES / 16, FOUT / 16), 32, 0, stream>>>(feat1, W2, feat2);

    attn_scores<<<cdiv((long)NODES, TB), TB, 0, stream>>>(feat2, al2, ar2, el2, er2, NODES, 1, FOUT);

    fill_f32<<<cdiv((long)NODES, TB), TB, 0, stream>>>(m2, NODES, -INFINITY);
    fill_f32<<<cdiv((long)NODES, TB), TB, 0, stream>>>(z2, NODES, 0.f);
    fill_f32<<<cdiv((long)NODES * FOUT, TB), TB, 0, stream>>>(out, (long)NODES * FOUT, 0.f);

    edge_max<<<cdiv((long)NEDGE, TB), TB, 0, stream>>>(src, dst, el2, er2, m2, NEDGE, 1);
    edge_expsum<<<cdiv((long)NEDGE, TB), TB, 0, stream>>>(src, dst, el2, er2, m2, z2, NEDGE, 1);
    edge_aggregate<<<cdiv((long)NEDGE, TB), TB, 0, stream>>>(src, dst, el2, er2, m2, z2,
                                                             feat2, out, NEDGE, 1, FOUT);

    bias_add<<<cdiv((long)NODES * FOUT, TB), TB, 0, stream>>>(out, b2, (long)NODES * FOUT, FOUT);
}